// GCN_46755013984832
// MI455X (gfx1250) — compile-verified
//
#include <hip/hip_runtime.h>
#include <hip/hip_bf16.h>

typedef __attribute__((ext_vector_type(2))) float v2f;
typedef __attribute__((ext_vector_type(8))) float v8f;

#define IN_F  128
#define H_F   128
#define OUT_F 3
#define BN_EPS 1e-5f

// ---------------------------------------------------------------------------
// Init: zero agg + stat accumulators, deg starts at 1.0 (self-loop weight).
// ---------------------------------------------------------------------------
__global__ __launch_bounds__(256) void k_init(float* __restrict__ agg,
                                              float* __restrict__ deg,
                                              float* __restrict__ sums,
                                              long long aggN, int n) {
  long long idx = (long long)blockIdx.x * blockDim.x + threadIdx.x;
  long long stride = (long long)gridDim.x * blockDim.x;
  for (long long i = idx; i < aggN; i += stride) agg[i] = 0.0f;
  for (long long i = idx; i < n; i += stride) deg[i] = 1.0f;
  if (idx < 256) sums[idx] = 0.0f;
}

// ---------------------------------------------------------------------------
// h = x @ conv_w  via V_WMMA_F32_16X16X4_F32.
// One 16x16 D tile per wave; 8 waves/block cover all 8 N-tiles of one M-tile.
// A 16x4 layout: lane L -> M = L%16, K pair = (L/16)*2 (+0,+1 in v[0],v[1]).
// B 4x16 layout: lane L -> N = L%16, K pair = (L/16)*2 (rows of conv_w).
// D layout: VGPR v, lane L -> M = v + 8*(L/16), N = L%16.
// ---------------------------------------------------------------------------
__global__ __launch_bounds__(256) void k_gemm(const float* __restrict__ x,
                                              const float* __restrict__ w,
                                              float* __restrict__ h, int n) {
  const int lane  = threadIdx.x & 31;
  const int wave  = threadIdx.x >> 5;       // 0..7 -> N tile
  const int tileM = blockIdx.x;
  const int mRow  = tileM * 16 + (lane & 15);
  const int m     = mRow < n ? mRow : n - 1;  // clamp keeps EXEC all-1s
  const int kHalf = (lane >> 4) * 2;          // 0 or 2
  const int nCol  = wave * 16 + (lane & 15);

  v8f acc = {};
#pragma unroll 4
  for (int k = 0; k < IN_F; k += 4) {
    v2f a = *(const v2f*)(x + (size_t)m * IN_F + k + kHalf);
    const float* bp = w + (size_t)(k + kHalf) * H_F + nCol;
    v2f b;
    b.x = bp[0];
    b.y = bp[H_F];
    acc = __builtin_amdgcn_wmma_f32_16x16x4_f32(
        /*neg_a=*/false, a, /*neg_b=*/false, b,
        /*c_mod=*/(short)0, acc, /*reuse_a=*/false, /*reuse_b=*/false);
  }

  const int rowBase = tileM * 16 + 8 * (lane >> 4);
  float* hp = h + (size_t)rowBase * H_F + wave * 16 + (lane & 15);
  if (tileM * 16 + 16 <= n) {
    // Full tile (always taken for n % 16 == 0): 8 unguarded, clause-able stores.
#pragma unroll
    for (int v = 0; v < 8; ++v) hp[(size_t)v * H_F] = acc[v];
  } else {
    // Cold ragged-edge fallback.
#pragma unroll
    for (int v = 0; v < 8; ++v) {
      if (rowBase + v < n) hp[(size_t)v * H_F] = acc[v];
    }
  }
}

// ---------------------------------------------------------------------------
// deg[col] += ew  (self-loop "1" already in init)
// ---------------------------------------------------------------------------
__global__ __launch_bounds__(256) void k_degree(const int* __restrict__ col,
                                                const float* __restrict__ ew,
                                                float* __restrict__ deg, int E) {
  int e = blockIdx.x * blockDim.x + threadIdx.x;
  if (e < E) atomicAdd(&deg[col[e]], ew[e]);
}

__global__ __launch_bounds__(256) void k_dinv(const float* __restrict__ deg,
                                              float* __restrict__ dinv, int n) {
  int i = blockIdx.x * blockDim.x + threadIdx.x;
  if (i < n) {
    float d = deg[i];
    dinv[i] = d > 0.0f ? rsqrtf(d) : 0.0f;
  }
}

// ---------------------------------------------------------------------------
// Message + scatter-add: one wave per (edge or self-loop), lane -> float4.
// h and agg are L2-resident (25.6 MB each << 192 MB), so the gathers and the
// float atomics resolve at L2 speed; this kernel bounds total runtime.
// ---------------------------------------------------------------------------
__global__ __launch_bounds__(256) void k_scatter(const int* __restrict__ ei,
                                                 const float* __restrict__ ew,
                                                 const float* __restrict__ dinv,
                                                 const float* __restrict__ h,
                                                 float* __restrict__ agg,
                                                 int E, int n) {
  const int lane = threadIdx.x & 31;
  long long e = (long long)blockIdx.x * (blockDim.x >> 5) + (threadIdx.x >> 5);
  if (e >= (long long)E + n) return;

  int r, c;
  float wgt;
  if (e < E) {
    r = ei[e];               // row
    c = ei[(size_t)E + e];   // col
    wgt = ew[e];
  } else {
    r = c = (int)(e - E);    // self loop
    wgt = 1.0f;
  }
  const float norm = dinv[r] * wgt * dinv[c];

  const float4 hv = *(const float4*)(h + (size_t)r * H_F + lane * 4);
  float* ap = agg + (size_t)c * H_F + lane * 4;
  atomicAdd(ap + 0, hv.x * norm);
  atomicAdd(ap + 1, hv.y * norm);
  atomicAdd(ap + 2, hv.z * norm);
  atomicAdd(ap + 3, hv.w * norm);
}

// ---------------------------------------------------------------------------
// h2 = relu(agg + bias) in place; per-channel sum / sumsq accumulation.
// blockDim=256 and grid stride are multiples of 128 -> idx%128 == tid%128.
// ---------------------------------------------------------------------------
__global__ __launch_bounds__(256) void k_stats(float* __restrict__ agg,
                                               const float* __restrict__ conv_b,
                                               float* __restrict__ sums, int n) {
  const int c = threadIdx.x & (H_F - 1);
  const float bias = conv_b[c];
  float s = 0.0f, s2 = 0.0f;
  const long long total = (long long)n * H_F;
  const long long stride = (long long)gridDim.x * blockDim.x;
  for (long long idx = (long long)blockIdx.x * blockDim.x + threadIdx.x;
       idx < total; idx += stride) {
    float v = agg[idx] + bias;
    v = fmaxf(v, 0.0f);
    agg[idx] = v;
    s += v;
    s2 += v * v;
  }
  atomicAdd(&sums[c], s);
  atomicAdd(&sums[H_F + c], s2);
}

// ---------------------------------------------------------------------------
// Fold BN into final linear:  effW[c][o] = scale[c]*lin_w[c][o]
//                             effB[o]    = lin_b[o] + sum_c shift[c]*lin_w[c][o]
// ---------------------------------------------------------------------------
__global__ __launch_bounds__(128) void k_finalize(const float* __restrict__ sums,
                                                  const float* __restrict__ gamma,
                                                  const float* __restrict__ beta,
                                                  const float* __restrict__ lin_w,
                                                  const float* __restrict__ lin_b,
                                                  float* __restrict__ effW,
                                                  float* __restrict__ effB, int n) {
  __shared__ float red[H_F];
  const int c = threadIdx.x;
  const float inv_n = 1.0f / (float)n;
  const float mean = sums[c] * inv_n;
  const float var = sums[H_F + c] * inv_n - mean * mean;
  const float scale = gamma[c] * rsqrtf(var + BN_EPS);
  const float shift = beta[c] - mean * scale;

  for (int o = 0; o < OUT_F; ++o) {
    effW[c * OUT_F + o] = scale * lin_w[c * OUT_F + o];
    red[c] = shift * lin_w[c * OUT_F + o];
    __syncthreads();
    for (int s = H_F / 2; s > 0; s >>= 1) {
      if (c < s) red[c] += red[c + s];
      __syncthreads();
    }
    if (c == 0) effB[o] = lin_b[o] + red[0];
    __syncthreads();
  }
}

// ---------------------------------------------------------------------------
// out[i][o] = sum_c h2[i][c]*effW[c][o] + effB[o]; wave per node, shuffle reduce.
// ---------------------------------------------------------------------------
__global__ __launch_bounds__(256) void k_linear(const float* __restrict__ h2,
                                                const float* __restrict__ effW,
                                                const float* __restrict__ effB,
                                                float* __restrict__ out, int n) {
  const int lane = threadIdx.x & 31;
  const int node = blockIdx.x * (blockDim.x >> 5) + (threadIdx.x >> 5);
  if (node >= n) return;

  const float4 hv = *(const float4*)(h2 + (size_t)node * H_F + lane * 4);
  const float hvv[4] = {hv.x, hv.y, hv.z, hv.w};
  float a0 = 0.0f, a1 = 0.0f, a2 = 0.0f;
  const int c0 = lane * 4;
#pragma unroll
  for (int j = 0; j < 4; ++j) {
    const float v = hvv[j];
    a0 += v * effW[(c0 + j) * OUT_F + 0];
    a1 += v * effW[(c0 + j) * OUT_F + 1];
    a2 += v * effW[(c0 + j) * OUT_F + 2];
  }
#pragma unroll
  for (int off = 16; off > 0; off >>= 1) {
    a0 += __shfl_down(a0, off, 32);
    a1 += __shfl_down(a1, off, 32);
    a2 += __shfl_down(a2, off, 32);
  }
  if (lane == 0) {
    out[(size_t)node * OUT_F + 0] = a0 + effB[0];
    out[(size_t)node * OUT_F + 1] = a1 + effB[1];
    out[(size_t)node * OUT_F + 2] = a2 + effB[2];
  }
}

// ---------------------------------------------------------------------------
extern "C" void kernel_launch(void* const* d_in, const int* in_sizes, int n_in,
                              void* d_out, int out_size, void* d_ws, size_t ws_size,
                              hipStream_t stream) {
  const float* x      = (const float*)d_in[0];
  const int*   ei     = (const int*)d_in[1];   // [2, E] flat: rows then cols
  const float* ew     = (const float*)d_in[2];
  const float* conv_w = (const float*)d_in[3];
  const float* conv_b = (const float*)d_in[4];
  const float* gamma  = (const float*)d_in[5];
  const float* beta   = (const float*)d_in[6];
  const float* lin_w  = (const float*)d_in[7];
  const float* lin_b  = (const float*)d_in[8];
  float* out = (float*)d_out;

  const int n = in_sizes[0] / IN_F;   // 50000
  const int E = in_sizes[1] / 2;      // 800000

  // Workspace layout (floats)
  float* ws   = (float*)d_ws;
  float* h    = ws;                         // n*128
  float* agg  = h + (size_t)n * H_F;        // n*128 (becomes h2 in place)
  float* deg  = agg + (size_t)n * H_F;      // n
  float* dinv = deg + n;                    // n
  float* sums = dinv + n;                   // 256 (sum | sumsq)
  float* effW = sums + 256;                 // 128*3
  float* effB = effW + H_F * OUT_F;         // 3

  k_init<<<1024, 256, 0, stream>>>(agg, deg, sums, (long long)n * H_F, n);

  const int mTiles = (n + 15) / 16;
  k_gemm<<<mTiles, 256, 0, stream>>>(x, conv_w, h, n);

  k_degree<<<(E + 255) / 256, 256, 0, stream>>>(ei + E, ew, deg, E);
  k_dinv<<<(n + 255) / 256, 256, 0, stream>>>(deg, dinv, n);

  const long long msgs = (long long)E + n;
  const int wavesPerBlock = 256 / 32;
  k_scatter<<<(int)((msgs + wavesPerBlock - 1) / wavesPerBlock), 256, 0, stream>>>(
      ei, ew, dinv, h, agg, E, n);

  k_stats<<<512, 256, 0, stream>>>(agg, conv_b, sums, n);
  k_finalize<<<1, 128, 0, stream>>>(sums, gamma, beta, lin_w, lin_b, effW, effB, n);
  k_linear<<<(n + wavesPerBlock - 1) / wavesPerBlock, 256, 0, stream>>>(
      agg, effW, effB, out, n);
}